// ChebyFilter_13632226197599
// MI455X (gfx1250) — compile-verified
//
#include <hip/hip_runtime.h>
#include <cmath>

#define N_USERS 100000
#define N_ITEMS 50000
#define NNZ_CT  1600000
#define BATCH   64
#define ORDER   8
#define FLATNESS 2

typedef float v2f __attribute__((ext_vector_type(2)));
typedef float v8f __attribute__((ext_vector_type(8)));

// ---------------- elementwise helpers ----------------

__global__ void fill_zero_k(float* p, int n) {
  int g = blockIdx.x * blockDim.x + threadIdx.x;
  if (g < n) p[g] = 0.0f;
}

// dst = sa*a + sb*b   (dst may alias a or b at identical indices)
__global__ void lincomb_k(float* dst, const float* a, float sa,
                          const float* b, float sb, int n) {
  int g = blockIdx.x * blockDim.x + threadIdx.x;
  if (g < n) dst[g] = sa * a[g] + sb * b[g];
}

// ---------------- LDS-tiled transpose: src[R][C] -> dst[C][R] ----------------
// Global->LDS staging uses the CDNA5 async copy path (GLOBAL_LOAD_ASYNC_TO_LDS),
// tracked by ASYNCcnt; per-lane EXEC masking handles partial tiles.
// NOTE: the LDS operand is derived from the real pointer (ptrtoint) so `tile`
// escapes into the asm and the "memory" clobber keeps the ds_load reads live.

__global__ void transpose_k(const float* __restrict__ src, float* __restrict__ dst,
                            int R, int C) {
  __shared__ float tile[32][33];
  int cb = blockIdx.x * 32, rb = blockIdx.y * 32;
  int tx = threadIdx.x, ty = threadIdx.y;  // block (32, 8)
  #pragma unroll
  for (int j = 0; j < 32; j += 8) {
    int r = rb + ty + j, c = cb + tx;
    if (r < R && c < C) {
      // low 32 bits of a generic LDS pointer = wave-relative LDS byte offset,
      // which is exactly what VDST of global_load_async_to_lds_b32 expects.
      unsigned lds_off = (unsigned)(uintptr_t)&tile[ty + j][tx];
      unsigned long long gaddr =
          (unsigned long long)(const void*)&src[(size_t)r * C + c];
      asm volatile("global_load_async_to_lds_b32 %0, %1, off"
                   :: "v"(lds_off), "v"(gaddr) : "memory");
    }
  }
  asm volatile("s_wait_asynccnt 0" ::: "memory");
  __syncthreads();
  #pragma unroll
  for (int j = 0; j < 32; j += 8) {
    int c = cb + ty + j, r = rb + tx;
    if (r < R && c < C) dst[(size_t)c * R + r] = tile[tx][ty + j];
  }
}

// ---------------- COO scatter: dst[si[e],:] += f*vals[e]*src[gi[e],:] --------
// One wave32 per edge: lane covers batch pair (2*lane, 2*lane+1) => full 64-wide
// row per wave (coalesced 256B gather + 2 f32 atomics per lane into L2-resident
// accumulators). Edge index is forced into an SGPR via readfirstlane so the
// row/col/vals stream lowers to scalar (SMEM) loads; next edge is prefetched.

__global__ void scatter_k(const int* __restrict__ gi, const int* __restrict__ si,
                          const float* __restrict__ vals,
                          const float* __restrict__ src,
                          float* __restrict__ dst, float f, int nnz) {
  int lane   = threadIdx.x & 31;
  int wave   = (blockIdx.x * blockDim.x + threadIdx.x) >> 5;
  int nwaves = (gridDim.x * blockDim.x) >> 5;
  int e0 = __builtin_amdgcn_readfirstlane(wave);   // wave-uniform in SGPR
  for (int e = e0; e < nnz; e += nwaves) {
    int en = e + nwaves;
    if (en < nnz) {
      __builtin_prefetch(&gi[en], 0, 1);
      __builtin_prefetch(&si[en], 0, 1);
      __builtin_prefetch(&vals[en], 0, 1);
    }
    int g = gi[e], s = si[e];
    float v = f * vals[e];
    const float2 xv = *(const float2*)&src[(size_t)g * BATCH + lane * 2];
    float* d = &dst[(size_t)s * BATCH + lane * 2];
    __hip_atomic_fetch_add(&d[0], v * xv.x, __ATOMIC_RELAXED, __HIP_MEMORY_SCOPE_AGENT);
    __hip_atomic_fetch_add(&d[1], v * xv.y, __ATOMIC_RELAXED, __HIP_MEMORY_SCOPE_AGENT);
  }
}

// ---------------- out_tile += c * T_tile via V_WMMA_F32_16X16X4_F32 ----------
// D = sum_{mb=0..3} (c * I[:,4mb:4mb+4]) x T[4mb:4mb+4, :] + C  ==  c*T + C.
// A layout (ISA): lanes0-15 M=0-15 {V0=K0,V1=K1}; lanes16-31 M=0-15 {K2,K3}.
// B layout (per 64x16 B table pattern): V0={K0|K2}, V1={K1|K3}, N=lane%16.
// C/D layout (ISA): VGPR v -> M=v (lanes0-15) / M=v+8 (lanes16-31), N=lane%16.

__global__ void axpy_wmma_k(float* out, const float* __restrict__ t, float c) {
  const int n_tiles = (N_ITEMS / 16) * (BATCH / 16);  // 3125 * 4, exact
  int tile = blockIdx.x * (blockDim.x >> 5) + (threadIdx.x >> 5);
  if (tile >= n_tiles) return;  // wave-uniform: EXEC stays all-ones for WMMA
  int lane = threadIdx.x & 31;
  int i0 = (tile >> 2) * 16;      // item block
  int b0 = (tile & 3) * 16;       // batch block
  int n = lane & 15, hi = lane >> 4;

  v8f acc;
  #pragma unroll
  for (int v = 0; v < 8; ++v)
    acc[v] = out[(size_t)(i0 + v + hi * 8) * BATCH + b0 + n];

  #pragma unroll
  for (int mb = 0; mb < 4; ++mb) {
    int k0 = mb * 4 + hi * 2;     // this lane-half's K indices: k0, k0+1
    v2f a, b;
    a.x = (n == k0)     ? c : 0.0f;   // A = c * I slab
    a.y = (n == k0 + 1) ? c : 0.0f;
    b.x = t[(size_t)(i0 + k0)     * BATCH + b0 + n];
    b.y = t[(size_t)(i0 + k0 + 1) * BATCH + b0 + n];
    acc = __builtin_amdgcn_wmma_f32_16x16x4_f32(false, a, false, b,
                                                (short)0, acc, false, false);
  }

  #pragma unroll
  for (int v = 0; v < 8; ++v)
    out[(size_t)(i0 + v + hi * 8) * BATCH + b0 + n] = acc[v];
}

// ---------------- host: replicate cheby_coeffs(8, 2) in double ----------------

static inline double round3(double v) { return std::nearbyint(v * 1000.0) / 1000.0; }

static void cheby_coeffs_host(float* cf) {
  const int order = ORDER;
  double tgt[ORDER + 1], nodes[ORDER + 1];
  for (int j = 0; j <= order; ++j) {
    double xv = round3(std::cos((double)(order - j) / order * M_PI));
    double tv = (xv < 0.0) ? std::pow(-xv, (double)FLATNESS) * 0.5 + 0.5
                           : std::pow(xv, (double)FLATNESS) * (-0.5) + 0.5;
    tgt[j] = round3(tv);
  }
  for (int j = 0; j <= order; ++j) {
    int k = j + 1;
    nodes[j] = std::cos(((double)order + 1.0 + 0.5 - (double)k) / (double)(order + 1) * M_PI);
  }
  double prev[ORDER + 1], cur[ORDER + 1], nxt[ORDER + 1], c[ORDER + 1];
  double s0 = 0.0, s1 = 0.0;
  for (int j = 0; j <= order; ++j) { prev[j] = tgt[j]; s0 += prev[j]; }
  for (int j = 0; j <= order; ++j) { cur[j] = nodes[j] * tgt[j]; s1 += cur[j]; }
  c[0] = s0; c[1] = s1;
  for (int m = 2; m <= order; ++m) {
    double sm = 0.0;
    for (int j = 0; j <= order; ++j) {
      nxt[j] = 2.0 * nodes[j] * cur[j] - prev[j];
      sm += nxt[j];
    }
    c[m] = sm;
    for (int j = 0; j <= order; ++j) { prev[j] = cur[j]; cur[j] = nxt[j]; }
  }
  for (int m = 0; m <= order; ++m) c[m] *= 2.0 / (double)(order + 1);
  c[0] /= 2.0;
  for (int m = 0; m <= order; ++m) cf[m] = (float)c[m];
}

// ---------------- launch ----------------

extern "C" void kernel_launch(void* const* d_in, const int* in_sizes, int n_in,
                              void* d_out, int out_size, void* d_ws, size_t ws_size,
                              hipStream_t stream) {
  const float* signal = (const float*)d_in[0];   // [64, 50000]
  const float* vals   = (const float*)d_in[1];   // [NNZ]
  const int*   row    = (const int*)d_in[2];     // [NNZ]
  const int*   col    = (const int*)d_in[3];     // [NNZ]
  float* out = (float*)d_out;                    // [64, 50000]

  float cf[ORDER + 1];
  cheby_coeffs_host(cf);

  const int SZ  = N_ITEMS * BATCH;   // 3.2M floats
  const int YSZ = N_USERS * BATCH;   // 6.4M floats
  float* t0   = (float*)d_ws;
  float* t1   = t0 + SZ;
  float* oacc = t1 + SZ;
  float* y    = oacc + SZ;           // total 16M floats = 64 MB in d_ws

  dim3 tb(32, 8);
  const int EW = 256;                                  // elementwise block
  const int axpy_blocks = ((N_ITEMS / 16) * (BATCH / 16) + 7) / 8;
  const int scat_blocks = 2048;                        // 16384 waves over 1.6M edges

  // t0 = signal^T  ([50000,64])
  transpose_k<<<dim3((N_ITEMS + 31) / 32, (BATCH + 31) / 32), tb, 0, stream>>>(
      signal, t0, BATCH, N_ITEMS);

  // out_acc = c0 * t0
  fill_zero_k<<<(SZ + EW - 1) / EW, EW, 0, stream>>>(oacc, SZ);
  axpy_wmma_k<<<axpy_blocks, 256, 0, stream>>>(oacc, t0, cf[0]);

  // t1 = lap(t0) = t0 - 2 * A^T (A t0)
  fill_zero_k<<<(YSZ + EW - 1) / EW, EW, 0, stream>>>(y, YSZ);
  scatter_k<<<scat_blocks, 256, 0, stream>>>(col, row, vals, t0, y, 1.0f, NNZ_CT);
  lincomb_k<<<(SZ + EW - 1) / EW, EW, 0, stream>>>(t1, t0, 1.0f, t0, 0.0f, SZ);
  scatter_k<<<scat_blocks, 256, 0, stream>>>(row, col, vals, y, t1, -2.0f, NNZ_CT);
  axpy_wmma_k<<<axpy_blocks, 256, 0, stream>>>(oacc, t1, cf[1]);

  // t2 = 2*lap(t1) - t0 ; out_acc += c[k]*t2 ; shift
  for (int k = 2; k <= ORDER; ++k) {
    fill_zero_k<<<(YSZ + EW - 1) / EW, EW, 0, stream>>>(y, YSZ);
    scatter_k<<<scat_blocks, 256, 0, stream>>>(col, row, vals, t1, y, 1.0f, NNZ_CT);
    lincomb_k<<<(SZ + EW - 1) / EW, EW, 0, stream>>>(t0, t1, 2.0f, t0, -1.0f, SZ);
    scatter_k<<<scat_blocks, 256, 0, stream>>>(row, col, vals, y, t0, -4.0f, NNZ_CT);
    axpy_wmma_k<<<axpy_blocks, 256, 0, stream>>>(oacc, t0, cf[k]);
    float* tmp = t0; t0 = t1; t1 = tmp;   // t0 <- old t1, t1 <- t2
  }

  // d_out = out_acc^T ([64, 50000])
  transpose_k<<<dim3((BATCH + 31) / 32, (N_ITEMS + 31) / 32), tb, 0, stream>>>(
      oacc, out, N_ITEMS, BATCH);
}